// ImMatchNet_29274497089745
// MI455X (gfx1250) — compile-verified
//
#include <hip/hip_runtime.h>

#define BD   4
#define CC   1024
#define HW   625        // 25*25
#define PPAD 640        // HW padded to multiple of 16
#define POS  390625     // 625^2 spatial positions of the 4D tensor
#define PADPOS 531441   // 27^4 zero-padded spatial positions
#define TAPP 84         // 3^4 = 81 taps padded to multiple of 4
#define EPS_N 1e-6f
#define EPS_M 1e-5f
#define NEG_BIG -3.402823466e38f

typedef __attribute__((ext_vector_type(2))) float v2f;
typedef __attribute__((ext_vector_type(8))) float v8f;

// -------------------- l2 norm (1/sqrt(sum_c x^2 + 1e-6)) --------------------
__global__ void invnorm_kernel(const float* __restrict__ f, float* __restrict__ invn) {
  int p = blockIdx.x, b = blockIdx.y, tid = threadIdx.x;
  __shared__ float red[256];
  float s = 0.f;
  for (int c = tid; c < CC; c += 256) {
    float v = f[(b * CC + c) * HW + p];
    s += v * v;
  }
  red[tid] = s; __syncthreads();
  for (int k = 128; k > 0; k >>= 1) {
    if (tid < k) red[tid] += red[tid + k];
    __syncthreads();
  }
  if (tid == 0) invn[b * HW + p] = rsqrtf(red[0] + EPS_N);
}

// fAn[b][p][c] = fA[b][c][p] * invA[b][p]   (p padded to 640 with zeros)
__global__ void packA_kernel(const float* __restrict__ fA, const float* __restrict__ invA,
                             float* __restrict__ fAn) {
  int idx = blockIdx.x * blockDim.x + threadIdx.x;
  if (idx >= BD * PPAD * CC) return;
  int c = idx % CC;
  int p = (idx / CC) % PPAD;
  int b = idx / (CC * PPAD);
  float v = 0.f;
  if (p < HW) v = fA[(b * CC + c) * HW + p] * invA[b * HW + p];
  fAn[idx] = v;
}

// pair-packed B: fBp[((b*512 + c/2)*640 + q)*2 + (c&1)] = fB[b][c][q]*invB[b][q]
__global__ void packB_kernel(const float* __restrict__ fB, const float* __restrict__ invB,
                             float* __restrict__ fBp) {
  int idx = blockIdx.x * blockDim.x + threadIdx.x;
  if (idx >= BD * CC * PPAD) return;
  int v  = idx & 1;
  int q  = (idx >> 1) % PPAD;
  int pr = ((idx >> 1) / PPAD) % 512;
  int b  = idx / (2 * PPAD * 512);
  int c  = pr * 2 + v;
  float val = 0.f;
  if (q < HW) val = fB[(b * CC + c) * HW + q] * invB[b * HW + q];
  fBp[idx] = val;
}

// ---------------- correlation GEMM: corr[b][p][q] = sum_c A[p][c]*B[c][q] ---
// fused epilogue: relu + x/sqrt(x^2+1e-6). One wave -> one 16x16 tile.
__global__ __launch_bounds__(32) void corr_wmma_kernel(
    const float* __restrict__ fAn, const float* __restrict__ fBp,
    float* __restrict__ corr) {
  int b = blockIdx.y;
  int tileq = blockIdx.x % 40;
  int tilep = blockIdx.x / 40;
  int lane = threadIdx.x & 31;
  int n = lane & 15;      // A row (M) and B col (N) for this lane
  int hi = lane >> 4;     // half-wave selects K sub-slice

  const float* Ab = fAn + (size_t)b * PPAD * CC + (size_t)(tilep * 16 + n) * CC + 2 * hi;
  const float* Bb = fBp + (size_t)b * 512 * PPAD * 2 + (size_t)(tileq * 16 + n) * 2 +
                    (size_t)hi * PPAD * 2;

  v8f acc = {};
  for (int k = 0; k < CC; k += 4) {
    v2f a  = *(const v2f*)(Ab + k);                          // A[m][k+2hi], A[m][k+2hi+1]
    v2f bb = *(const v2f*)(Bb + (size_t)(k >> 1) * PPAD * 2); // B[k+2hi][n], B[k+2hi+1][n]
    acc = __builtin_amdgcn_wmma_f32_16x16x4_f32(false, a, false, bb,
                                                (short)0, acc, false, false);
  }
  int q = tileq * 16 + n;
  if (q < HW) {
#pragma unroll
    for (int r = 0; r < 8; ++r) {
      int p = tilep * 16 + r + 8 * hi;
      if (p < HW) {
        float v = fmaxf(acc[r], 0.f);
        v = v * rsqrtf(v * v + EPS_N);
        corr[((size_t)b * HW + p) * HW + q] = v;
      }
    }
  }
}

// -------------------- mutual matching helpers --------------------
__global__ void maxq_kernel(const float* __restrict__ x, float* __restrict__ mx) {
  int p = blockIdx.x, b = blockIdx.y, tid = threadIdx.x;
  __shared__ float red[256];
  const float* row = x + ((size_t)b * HW + p) * HW;
  float m = NEG_BIG;
  for (int q = tid; q < HW; q += 256) m = fmaxf(m, row[q]);
  red[tid] = m; __syncthreads();
  for (int k = 128; k > 0; k >>= 1) {
    if (tid < k) red[tid] = fmaxf(red[tid], red[tid + k]);
    __syncthreads();
  }
  if (tid == 0) mx[b * HW + p] = red[0];
}

__global__ void maxp_kernel(const float* __restrict__ x, float* __restrict__ mx) {
  int q = blockIdx.x * blockDim.x + threadIdx.x;
  int b = blockIdx.y;
  if (q >= HW) return;
  float m = NEG_BIG;
  for (int p = 0; p < HW; ++p) m = fmaxf(m, x[((size_t)b * HW + p) * HW + q]);
  mx[b * HW + q] = m;
}

__global__ void mutual_kernel(const float* __restrict__ x, const float* __restrict__ mxA,
                              const float* __restrict__ mxB, float* __restrict__ out) {
  int idx = blockIdx.x * blockDim.x + threadIdx.x;
  if (idx >= BD * HW * HW) return;
  int q = idx % HW;
  int p = (idx / HW) % HW;
  int b = idx / (HW * HW);
  float v = x[idx];
  out[idx] = v * (v / (mxA[b * HW + p] + EPS_M)) * (v / (mxB[b * HW + q] + EPS_M));
}

// out = ya + transpose(yb)
__global__ void addT_kernel(const float* __restrict__ ya, const float* __restrict__ yb,
                            float* __restrict__ out) {
  int idx = blockIdx.x * blockDim.x + threadIdx.x;
  if (idx >= BD * HW * HW) return;
  int q = idx % HW;
  int p = (idx / HW) % HW;
  int b = idx / (HW * HW);
  out[idx] = ya[idx] + yb[((size_t)b * HW + q) * HW + p];
}

// -------------------- padded-layout helpers --------------------
__global__ void fill0_kernel(float* __restrict__ p, int n) {
  int idx = blockIdx.x * blockDim.x + threadIdx.x;
  if (idx < n) p[idx] = 0.f;
}

// xp[b][27][27][27][27] <- x[b][625*625] (single channel), optional (f1f2)<->(f3f4) swap
__global__ void pad_kernel(const float* __restrict__ x, float* __restrict__ xp,
                           int do_transpose) {
  int idx = blockIdx.x * blockDim.x + threadIdx.x;
  if (idx >= BD * PADPOS) return;
  int b = idx / PADPOS;
  int t = idx % PADPOS;
  int j1 = t / 19683; t -= j1 * 19683;
  int j2 = t / 729;   t -= j2 * 729;
  int j3 = t / 27;
  int j4 = t - j3 * 27;
  float v = 0.f;
  if (j1 >= 1 && j1 <= 25 && j2 >= 1 && j2 <= 25 &&
      j3 >= 1 && j3 <= 25 && j4 >= 1 && j4 <= 25) {
    int i1 = j1 - 1, i2 = j2 - 1, i3 = j3 - 1, i4 = j4 - 1;
    int src = do_transpose ? (((i3 * 25 + i4) * 25 + i1) * 25 + i2)
                           : (((i1 * 25 + i2) * 25 + i3) * 25 + i4);
    v = x[(size_t)b * POS + src];
  }
  xp[idx] = v;
}

// -------------------- 4D conv via implicit-GEMM WMMA --------------------
// pair-packed weights: wpk[((ci*42 + t/2)*16 + n)*2 + (t&1)] = W[n][ci][tap t]
__global__ void packw_kernel(const float* __restrict__ w, float* __restrict__ wpk,
                             int cin, int cout) {
  int idx = blockIdx.x * blockDim.x + threadIdx.x;
  if (idx >= cin * 1344) return;              // 1344 = 42*16*2
  int vv = idx & 1;
  int n  = (idx >> 1) & 15;
  int pt = (idx >> 5) % 42;
  int ci = idx / 1344;
  int t  = pt * 2 + vv;
  float val = 0.f;
  if (n < cout && t < 81) val = w[(n * cin + ci) * 81 + t];
  wpk[idx] = val;
}

// M=16 padded-spatial positions, N=16 (cout padded), K=cin*84.
// Input is zero-padded 27^4 so the inner loop is branch-free:
//   ds_load_b64 (tap offsets) + ds_load_b64 (weight pair) + 2 global_load + wmma
__global__ __launch_bounds__(256) void conv4d_wmma_kernel(
    const float* __restrict__ xp, const float* __restrict__ wpk,
    const float* __restrict__ bias, float* __restrict__ y,
    int cin, int cout, int out_padded) {
  __shared__ float lw[10 * 1344];                 // <= 53.76 KB
  __shared__ __align__(8) int off_tab[TAPP];
  int tid = threadIdx.x;
  int wtot = cin * 1344;
  for (int i = tid; i < wtot; i += 256) lw[i] = wpk[i];
  if (tid < TAPP) {
    int t = tid, off = 0;
    if (t < 81) {
      int k1 = t / 27, tr = t - k1 * 27;
      int k2 = tr / 9; tr -= k2 * 9;
      int k3 = tr / 3, k4 = tr - k3 * 3;
      off = k1 * 19683 + k2 * 729 + k3 * 27 + k4;   // padded-layout tap offset
    }
    off_tab[t] = off;                                // taps 81..83 -> 0 (weights are 0)
  }
  __syncthreads();

  int wave = tid >> 5, lane = tid & 31;
  int b = blockIdx.y;
  int tile = blockIdx.x * 8 + wave;
  int p0 = tile * 16;
  int m  = lane & 15;
  int hi = lane >> 4;
  int p  = p0 + m;
  int pp = (p < POS) ? p : 0;                        // clamp; garbage rows never stored
  int i1 = pp / 15625; int r0 = pp - i1 * 15625;
  int i2 = r0 / 625;   r0 -= i2 * 625;
  int i3 = r0 / 25;
  int i4 = r0 - i3 * 25;
  int base = ((i1 * 27 + i2) * 27 + i3) * 27 + i4;   // top-left corner in padded layout

  v8f acc = {};
  for (int ci = 0; ci < cin; ++ci) {
    const float* xb  = xp + (size_t)(b * cin + ci) * PADPOS + base;
    const float* lwc = lw + ci * 1344;
    for (int t0 = 0; t0 < TAPP; t0 += 4) {
      int2 off = *(const int2*)&off_tab[t0 + 2 * hi];
      v2f bv = *(const v2f*)&lwc[((t0 >> 1) + hi) * 32 + m * 2];
      v2f av;
      av.x = xb[off.x];
      av.y = xb[off.y];
      acc = __builtin_amdgcn_wmma_f32_16x16x4_f32(false, av, false, bv,
                                                  (short)0, acc, false, false);
    }
  }

  int n = lane & 15;
  if (n < cout) {
    float bsv = bias[n];
#pragma unroll
    for (int r = 0; r < 8; ++r) {
      int pm = p0 + r + 8 * hi;
      if (pm < POS) {
        float v = fmaxf(acc[r] + bsv, 0.f);          // conv + bias + ReLU
        if (out_padded) {
          int a1 = pm / 15625; int t = pm - a1 * 15625;
          int a2 = t / 625;    t -= a2 * 625;
          int a3 = t / 25;
          int a4 = t - a3 * 25;
          size_t oidx = (size_t)(((a1 + 1) * 27 + (a2 + 1)) * 27 + (a3 + 1)) * 27 + (a4 + 1);
          y[(size_t)(b * cout + n) * PADPOS + oidx] = v;
        } else {
          y[(size_t)(b * cout + n) * POS + pm] = v;
        }
      }
    }
  }
}

// -------------------- host orchestration --------------------
extern "C" void kernel_launch(void* const* d_in, const int* in_sizes, int n_in,
                              void* d_out, int out_size, void* d_ws, size_t ws_size,
                              hipStream_t stream) {
  const float* fA = (const float*)d_in[0];
  const float* fB = (const float*)d_in[1];
  const float* W1 = (const float*)d_in[2];
  const float* b1 = (const float*)d_in[3];
  const float* W2 = (const float*)d_in[4];
  const float* b2 = (const float*)d_in[5];
  const float* W3 = (const float*)d_in[6];
  const float* b3 = (const float*)d_in[7];
  float* out = (float*)d_out;

  char* ws = (char*)d_ws;
  size_t off = 0;
  auto alloc = [&](size_t nElem) -> float* {
    float* p = (float*)(ws + off);
    off += nElem * sizeof(float);
    return p;
  };

  const size_t NCORR = (size_t)BD * HW * HW;        // 1,562,500
  const size_t NYP   = (size_t)BD * 10 * PADPOS;    // 21,257,640 (85 MB)
  const size_t NXP   = (size_t)BD * PADPOS;         // 2,125,764

  float* R1   = alloc(NYP);       // fAn|fBp early, y1p later (aliased)
  float* y2p  = alloc(NYP);
  float* fbuf = alloc(NCORR);     // corr, then final pre-mutual tensor
  float* x0   = alloc(NCORR);
  float* x0p  = alloc(NXP);
  float* xt0p = alloc(NXP);
  float* y3a  = alloc(NCORR);
  float* y3b  = alloc(NCORR);
  float* invA = alloc(BD * HW);
  float* invB = alloc(BD * HW);
  float* mxA  = alloc(BD * HW);
  float* mxB  = alloc(BD * HW);
  float* wpk1 = alloc(1  * 1344);
  float* wpk2 = alloc(10 * 1344);
  float* wpk3 = alloc(10 * 1344);

  float* fAn = R1;
  float* fBp = R1 + (size_t)BD * PPAD * CC;         // both fit under NYP region
  float* y1p = R1;                                   // reused after correlation

  // 1) feature L2 norms
  invnorm_kernel<<<dim3(HW, BD), 256, 0, stream>>>(fA, invA);
  invnorm_kernel<<<dim3(HW, BD), 256, 0, stream>>>(fB, invB);

  // 2) pack normalized features for WMMA GEMM
  int totP = BD * PPAD * CC;
  packA_kernel<<<(totP + 255) / 256, 256, 0, stream>>>(fA, invA, fAn);
  packB_kernel<<<(totP + 255) / 256, 256, 0, stream>>>(fB, invB, fBp);

  // 3) correlation GEMM (WMMA f32) with fused relu + l2norm epilogue
  corr_wmma_kernel<<<dim3(40 * 40, BD), 32, 0, stream>>>(fAn, fBp, fbuf);

  // 4) mutual matching #1
  int totC = (int)NCORR;
  maxq_kernel<<<dim3(HW, BD), 256, 0, stream>>>(fbuf, mxA);
  maxp_kernel<<<dim3((HW + 255) / 256, BD), 256, 0, stream>>>(fbuf, mxB);
  mutual_kernel<<<(totC + 255) / 256, 256, 0, stream>>>(fbuf, mxA, mxB, x0);

  // 5) zero-padded (and transposed) conv inputs
  int totXP = (int)NXP;
  pad_kernel<<<(totXP + 255) / 256, 256, 0, stream>>>(x0, x0p, 0);
  pad_kernel<<<(totXP + 255) / 256, 256, 0, stream>>>(x0, xt0p, 1);

  // 6) pack conv weights into pair-packed WMMA B-fragment layout
  packw_kernel<<<(1  * 1344 + 255) / 256, 256, 0, stream>>>(W1, wpk1, 1, 10);
  packw_kernel<<<(10 * 1344 + 255) / 256, 256, 0, stream>>>(W2, wpk2, 10, 10);
  packw_kernel<<<(10 * 1344 + 255) / 256, 256, 0, stream>>>(W3, wpk3, 10, 1);

  // 7) zero the padded intermediates (borders must be 0; interiors get overwritten)
  fill0_kernel<<<((int)NYP + 255) / 256, 256, 0, stream>>>(y1p, (int)NYP);
  fill0_kernel<<<((int)NYP + 255) / 256, 256, 0, stream>>>(y2p, (int)NYP);

  // 8) conv stack, branch 1 (x) and branch 2 (transposed x)
  int tiles = (POS + 15) / 16;                      // 24415
  dim3 cgrid((tiles + 7) / 8, BD);
  conv4d_wmma_kernel<<<cgrid, 256, 0, stream>>>(x0p,  wpk1, b1, y1p, 1, 10, 1);
  conv4d_wmma_kernel<<<cgrid, 256, 0, stream>>>(y1p,  wpk2, b2, y2p, 10, 10, 1);
  conv4d_wmma_kernel<<<cgrid, 256, 0, stream>>>(y2p,  wpk3, b3, y3a, 10, 1, 0);
  conv4d_wmma_kernel<<<cgrid, 256, 0, stream>>>(xt0p, wpk1, b1, y1p, 1, 10, 1);
  conv4d_wmma_kernel<<<cgrid, 256, 0, stream>>>(y1p,  wpk2, b2, y2p, 10, 10, 1);
  conv4d_wmma_kernel<<<cgrid, 256, 0, stream>>>(y2p,  wpk3, b3, y3b, 10, 1, 0);

  // 9) symmetric combine: f = S(x) + T(S(T(x)))
  addT_kernel<<<(totC + 255) / 256, 256, 0, stream>>>(y3a, y3b, fbuf);

  // 10) mutual matching #2 -> final output
  maxq_kernel<<<dim3(HW, BD), 256, 0, stream>>>(fbuf, mxA);
  maxp_kernel<<<dim3((HW + 255) / 256, BD), 256, 0, stream>>>(fbuf, mxB);
  mutual_kernel<<<(totC + 255) / 256, 256, 0, stream>>>(fbuf, mxA, mxB, out);
}